// KernelRBF_65481071397283
// MI455X (gfx1250) — compile-verified
//
#include <hip/hip_runtime.h>

typedef __attribute__((ext_vector_type(2))) float v2f;
typedef __attribute__((ext_vector_type(8))) float v8f;

#define N_ROWS 8192
#define M_ROWS 8192
#define DIM    512

#define BT  128           // block tile (rows of X and rows of Y)
#define KT  32            // K tile resident in LDS
#define LDS_STRIDE 36     // 32 + 4 pad: keeps 16B alignment, spreads banks

// ---------------------------------------------------------------------------
// CDNA5 async global->LDS DMA (bypasses VGPRs, tracked by ASYNCcnt).
// VDST = per-lane LDS byte address (low 32 bits of generic LDS pointer),
// VADDR = 64-bit global address, no SADDR.
// ---------------------------------------------------------------------------
__device__ __forceinline__ void async_copy_b128(const float* gptr, float* lptr) {
  const unsigned lds_addr = (unsigned)(size_t)lptr;
  asm volatile("global_load_async_to_lds_b128 %0, %1, off"
               :: "v"(lds_addr), "v"(gptr)
               : "memory");
}

__device__ __forceinline__ void wait_async_zero() {
  asm volatile("s_wait_asynccnt 0x0" ::: "memory");
}

// ---------------------------------------------------------------------------
// Pass 1: row squared-norms, one wave (32 lanes) per row.
// ---------------------------------------------------------------------------
__global__ __launch_bounds__(256) void rbf_norms_kernel(
    const float* __restrict__ v, float* __restrict__ out, int rows) {
  const int wave = blockIdx.x * (blockDim.x >> 5) + (threadIdx.x >> 5);
  const int lane = threadIdx.x & 31;
  if (wave >= rows) return;
  const float* p = v + (size_t)wave * DIM;
  float s = 0.0f;
  #pragma unroll
  for (int k = lane; k < DIM; k += 32) {
    float t = p[k];
    s += t * t;
  }
  #pragma unroll
  for (int off = 16; off > 0; off >>= 1)
    s += __shfl_xor(s, off, 32);
  if (lane == 0) out[wave] = s;
}

// ---------------------------------------------------------------------------
// Pass 2: tiled fp32 WMMA GEMM (X . Y^T) with fused RBF epilogue.
// Block: 256 threads = 8 waves, computes a 128x128 output tile.
// Waves arranged 4 (I) x 2 (J); each wave owns 32x64 = 2x4 tiles of 16x16.
// Global->LDS movement uses async DMA double-buffering.
// ---------------------------------------------------------------------------
__global__ __launch_bounds__(256) void rbf_wmma_gemm_kernel(
    const float* __restrict__ X, const float* __restrict__ Y,
    const float* __restrict__ x2, const float* __restrict__ y2,
    const float* __restrict__ gamma, float* __restrict__ out) {

  __shared__ float ldsX[2][BT * LDS_STRIDE];
  __shared__ float ldsY[2][BT * LDS_STRIDE];

  const int tid    = threadIdx.x;
  const int lane   = tid & 31;
  const int waveId = tid >> 5;
  const int wi     = waveId & 3;   // I sub-block (32 rows each)
  const int wj     = waveId >> 2;  // J sub-block (64 cols each)
  const int i0     = blockIdx.y * BT;
  const int j0     = blockIdx.x * BT;

  const int lrow  = lane & 15;     // M (or N) index within 16x16 fragment
  const int khalf = lane >> 4;     // selects K {0,1} vs K {2,3} pair

  v8f acc[2][4];
  #pragma unroll
  for (int a = 0; a < 2; ++a)
    #pragma unroll
    for (int b = 0; b < 4; ++b) {
      v8f z = {0.f, 0.f, 0.f, 0.f, 0.f, 0.f, 0.f, 0.f};
      acc[a][b] = z;
    }

  // Async tile loader: 128 rows x 32 K-floats per matrix, one b128 per
  // lane per chunk, DMA'd straight into LDS (no VGPR staging).
  auto load_tiles_async = [&](int kt, int buf) {
    const int k0 = kt * KT;
    #pragma unroll
    for (int it = 0; it < 4; ++it) {
      const int c   = tid + 256 * it;   // 0..1023 float4 chunks
      const int row = c >> 3;           // 0..127
      const int col = (c & 7) * 4;      // 0,4,...,28
      async_copy_b128(X + (size_t)(i0 + row) * DIM + k0 + col,
                      &ldsX[buf][row * LDS_STRIDE + col]);
      async_copy_b128(Y + (size_t)(j0 + row) * DIM + k0 + col,
                      &ldsY[buf][row * LDS_STRIDE + col]);
    }
  };

  load_tiles_async(0, 0);
  wait_async_zero();
  __syncthreads();

  const int nk = DIM / KT;  // 16 K-tiles
  for (int kt = 0; kt < nk; ++kt) {
    // Kick off next K-tile's DMA; it lands in the other buffer while the
    // WMMA block below runs (compute only waits on DScnt, not ASYNCcnt).
    if (kt + 1 < nk) load_tiles_async(kt + 1, (kt + 1) & 1);

    const float* bx = ldsX[kt & 1];
    const float* by = ldsY[kt & 1];

    #pragma unroll
    for (int ks = 0; ks < KT; ks += 4) {
      const int kk = ks + khalf * 2;  // per-lane K pair per ISA 16x4 layout
      v2f afrag[2];
      v2f bfrag[4];
      #pragma unroll
      for (int a = 0; a < 2; ++a)
        afrag[a] = *(const v2f*)&bx[(wi * 32 + a * 16 + lrow) * LDS_STRIDE + kk];
      #pragma unroll
      for (int b = 0; b < 4; ++b)
        bfrag[b] = *(const v2f*)&by[(wj * 64 + b * 16 + lrow) * LDS_STRIDE + kk];

      #pragma unroll
      for (int a = 0; a < 2; ++a)
        #pragma unroll
        for (int b = 0; b < 4; ++b)
          acc[a][b] = __builtin_amdgcn_wmma_f32_16x16x4_f32(
              /*neg_a=*/false, afrag[a], /*neg_b=*/false, bfrag[b],
              /*c_mod=*/(short)0, acc[a][b],
              /*reuse_a=*/false, /*reuse_b=*/false);
    }

    // Our async DMAs for the next buffer must have landed before any wave
    // starts reading it; every wave waits for its own, barrier covers all.
    wait_async_zero();
    __syncthreads();
  }

  // Fused RBF epilogue: out = exp(-g * (x2 + y2 - 2*dot))
  const float g = gamma[0];
  #pragma unroll
  for (int a = 0; a < 2; ++a) {
    const int ibase = i0 + wi * 32 + a * 16 + khalf * 8;  // + v gives the row
    #pragma unroll
    for (int b = 0; b < 4; ++b) {
      const int jcol = j0 + wj * 64 + b * 16 + lrow;
      const float yn = y2[jcol];
      #pragma unroll
      for (int v = 0; v < 8; ++v) {
        const int row = ibase + v;
        const float d = x2[row] + yn - 2.0f * acc[a][b][v];
        out[(size_t)row * M_ROWS + jcol] = __expf(-g * d);
      }
    }
  }
}

// ---------------------------------------------------------------------------
extern "C" void kernel_launch(void* const* d_in, const int* in_sizes, int n_in,
                              void* d_out, int out_size, void* d_ws, size_t ws_size,
                              hipStream_t stream) {
  const float* x     = (const float*)d_in[0];
  const float* y     = (const float*)d_in[1];
  const float* gamma = (const float*)d_in[2];
  float* out = (float*)d_out;

  float* x2 = (float*)d_ws;          // N floats
  float* y2 = x2 + N_ROWS;           // M floats

  rbf_norms_kernel<<<N_ROWS / 8, 256, 0, stream>>>(x, x2, N_ROWS);
  rbf_norms_kernel<<<M_ROWS / 8, 256, 0, stream>>>(y, y2, M_ROWS);

  dim3 grid(M_ROWS / BT, N_ROWS / BT);  // 64 x 64 blocks
  rbf_wmma_gemm_kernel<<<grid, 256, 0, stream>>>(x, y, x2, y2, gamma, out);
}